// AssocKernelAdapter_58729382805733
// MI455X (gfx1250) — compile-verified
//
#include <hip/hip_runtime.h>
#include <math.h>

#define TT 2048
#define EE 768
#define HH 12
#define DD 64

typedef __attribute__((ext_vector_type(2))) float v2f;
typedef __attribute__((ext_vector_type(8))) float v8f;

static __device__ __forceinline__ v8f wmma_f32(v2f a, v2f b, v8f c) {
  // D = A(16x4) * B(4x16) + C, full f32
  return __builtin_amdgcn_wmma_f32_16x16x4_f32(false, a, false, b, (short)0, c,
                                               false, false);
}

// ---------------------------------------------------------------- RMSNorm
__global__ void k_rmsnorm(const float* __restrict__ x, const float* __restrict__ w,
                          float* __restrict__ h) {
  const int t = blockIdx.x;
  const float* xr = x + (size_t)t * EE;
  float* hr = h + (size_t)t * EE;
  float ss = 0.f;
  for (int e = threadIdx.x; e < EE; e += 256) { float v = xr[e]; ss += v * v; }
  __shared__ float red[256];
  red[threadIdx.x] = ss;
  __syncthreads();
  for (int off = 128; off > 0; off >>= 1) {
    if (threadIdx.x < off) red[threadIdx.x] += red[threadIdx.x + off];
    __syncthreads();
  }
  const float r = rsqrtf(red[0] / (float)EE + 1e-6f);
  for (int e = threadIdx.x; e < EE; e += 256) hr[e] = xr[e] * r * w[e];
}

// ------------------------------------------- GEMM: C[M,N] = A[M,K] * B[N,K]^T
// one wave per 16x64 output tile, f32 WMMA 16x16x4.
// K processed in groups of 16 (20 loads + 16 WMMAs), register double-buffered
// so group n+1's loads are in flight while group n's WMMAs execute.
static __device__ __forceinline__ void gemm_load_group(const float* __restrict__ arow,
                                                       const float* __restrict__ brow,
                                                       int k0, int K, v2f a[4],
                                                       v2f b[4][4]) {
#pragma unroll
  for (int kk = 0; kk < 4; ++kk) a[kk] = *(const v2f*)(arow + k0 + kk * 4);
#pragma unroll
  for (int ng = 0; ng < 4; ++ng)
#pragma unroll
    for (int kk = 0; kk < 4; ++kk)
      b[ng][kk] = *(const v2f*)(brow + (size_t)(ng * 16) * K + k0 + kk * 4);
}

static __device__ __forceinline__ void gemm_mma_group(const v2f a[4],
                                                      const v2f b[4][4], v8f acc[4]) {
#pragma unroll
  for (int kk = 0; kk < 4; ++kk)
#pragma unroll
    for (int ng = 0; ng < 4; ++ng) acc[ng] = wmma_f32(a[kk], b[ng][kk], acc[ng]);
}

__global__ void k_gemm_nt(const float* __restrict__ A, const float* __restrict__ B,
                          float* __restrict__ C, int M, int N, int K) {
  const int n0 = blockIdx.x * 64;
  const int m0 = blockIdx.y * 16;
  const int lane = threadIdx.x;
  const int half = lane >> 4;
  const int lid = lane & 15;
  const int koff = half * 2;

  v8f acc[4] = {v8f{0.f}, v8f{0.f}, v8f{0.f}, v8f{0.f}};
  const float* arow = A + (size_t)(m0 + lid) * K + koff;
  const float* brow = B + (size_t)(n0 + lid) * K + koff;

  v2f a0[4], b0[4][4], a1[4], b1[4][4];
  gemm_load_group(arow, brow, 0, K, a0, b0);
  // K assumed multiple of 32 (768 = 24*32)
  for (int k0 = 0; k0 < K; k0 += 32) {
    gemm_load_group(arow, brow, k0 + 16, K, a1, b1);
    gemm_mma_group(a0, b0, acc);
    if (k0 + 32 < K) gemm_load_group(arow, brow, k0 + 32, K, a0, b0);
    gemm_mma_group(a1, b1, acc);
  }
#pragma unroll
  for (int ng = 0; ng < 4; ++ng)
#pragma unroll
    for (int r = 0; r < 8; ++r)
      C[(size_t)(m0 + r + 8 * half) * N + n0 + ng * 16 + lid] = acc[ng][r];
}

// -------------------------------- head split: L2-normalize q, transpose v
__global__ void k_headprep(const float* __restrict__ sh, const float* __restrict__ va,
                           float* __restrict__ qn, float* __restrict__ vT) {
  const int t = blockIdx.x;
  const int hh = blockIdx.y;
  const int lane = threadIdx.x;  // 32 threads, 2 elems each
  const float q0 = sh[(size_t)t * EE + hh * DD + lane];
  const float q1 = sh[(size_t)t * EE + hh * DD + lane + 32];
  float ss = q0 * q0 + q1 * q1;
#pragma unroll
  for (int mk = 16; mk > 0; mk >>= 1) ss += __shfl_xor(ss, mk, 32);
  const float inv = 1.f / fmaxf(sqrtf(ss), 1e-12f);
  qn[((size_t)hh * TT + t) * DD + lane] = q0 * inv;
  qn[((size_t)hh * TT + t) * DD + lane + 32] = q1 * inv;
  vT[((size_t)hh * DD + lane) * TT + t] = va[(size_t)t * EE + hh * DD + lane];
  vT[((size_t)hh * DD + lane + 32) * TT + t] = va[(size_t)t * EE + hh * DD + lane + 32];
}

// --------------------- emotion bias stats (rank-1 kernel => O(T^2) cheap)
__global__ void k_emo_partial(const float* __restrict__ e, float* __restrict__ pS,
                              float* __restrict__ pQ) {
  const int i = blockIdx.x * 256 + threadIdx.x;
  float S = 0.f, Q = 0.f;
  for (int j = 0; j < TT; ++j) {
    const int ad = (i >= j) ? (i - j) : (j - i);
    const float k = __expf(-(float)ad * 0.125f);
    S += k;
    Q += k * k;
  }
  pS[i] = e[i] * S;
  pQ[i] = e[i] * e[i] * Q;
}

__global__ void k_emo_reduce(const float* __restrict__ pS, const float* __restrict__ pQ,
                             float* __restrict__ stats) {
  __shared__ float rs[256], rq[256];
  float s = 0.f, q = 0.f;
  for (int i = threadIdx.x; i < TT; i += 256) { s += pS[i]; q += pQ[i]; }
  rs[threadIdx.x] = s;
  rq[threadIdx.x] = q;
  __syncthreads();
  for (int off = 128; off > 0; off >>= 1) {
    if (threadIdx.x < off) {
      rs[threadIdx.x] += rs[threadIdx.x + off];
      rq[threadIdx.x] += rq[threadIdx.x + off];
    }
    __syncthreads();
  }
  if (threadIdx.x == 0) {
    const float N = (float)TT * (float)TT;
    const float mean = rs[0] / N;
    const float var = (rq[0] - N * mean * mean) / (N - 1.f);
    stats[0] = mean;
    stats[1] = 1.f / fmaxf(sqrtf(var), 1e-6f);
  }
}

// ------------------------------------------------------------- attention
// batched B loads: 16 loads in flight, then a 16-WMMA accumulation chain.
static __device__ __forceinline__ v8f score_tile(const float* __restrict__ qrow_j,
                                                 const v2f afrag[16]) {
  v2f b[16];
#pragma unroll
  for (int kk = 0; kk < 16; ++kk) b[kk] = *(const v2f*)(qrow_j + kk * 4);
  v8f sc = {0.f};
#pragma unroll
  for (int kk = 0; kk < 16; ++kk) sc = wmma_f32(afrag[kk], b[kk], sc);
  return sc;
}

static __device__ __forceinline__ float bias_elem(float simraw, int i, int j,
                                                  float inv_s2, float gl, float gg,
                                                  float ei, float mean, float invstd) {
  const float sim = fminf(fmaxf(simraw, -1.f), 1.f);
  const float dist = fmaxf(2.f - 2.f * sim, 0.f);
  float s = fminf(fmaxf(-dist * inv_s2, -12.f), 12.f);
  const int ad = (i >= j) ? (i - j) : (j - i);
  const float gate =
      fmaxf(((ad <= 128) ? gl : 0.f) + (((j & 7) == 0) ? gg : 0.f), 1e-4f);
  s += __logf(gate);
  const float kern = __expf(-(float)ad * 0.125f);
  s += (ei * kern - mean) * invstd * 0.5f;
  return s;
}

__global__ void k_attn(const float* __restrict__ qn, const float* __restrict__ vT,
                       const float* __restrict__ emo, const float* __restrict__ stats,
                       const float* __restrict__ sigp, const float* __restrict__ blp,
                       const float* __restrict__ bgp, float* __restrict__ attn,
                       float* __restrict__ merged) {
  const int i0 = blockIdx.x * 16;
  const int hh = blockIdx.y;
  const int lane = threadIdx.x;
  const int half = lane >> 4;
  const int lid = lane & 15;
  const int koff = half * 2;

  const float sig = fmaxf(sigp[0], 1e-4f);
  const float inv_s2 = 1.f / (sig * sig);
  const float gl = 1.f / (1.f + __expf(-blp[0]));
  const float gg = 1.f / (1.f + __expf(-bgp[0]));
  const float mean = stats[0], invstd = stats[1];

  const float* qh = qn + (size_t)hh * TT * DD;
  const float* vh = vT + (size_t)hh * DD * TT;

  // A fragments (rows i0..i0+15) are invariant over all column tiles: preload.
  v2f afrag[16];
  {
    const float* ar = qh + (size_t)(i0 + lid) * DD + koff;
#pragma unroll
    for (int kk = 0; kk < 16; ++kk) afrag[kk] = *(const v2f*)(ar + kk * 4);
  }

  float er[8], mrun[8], srun[8];
#pragma unroll
  for (int r = 0; r < 8; ++r) {
    er[r] = emo[i0 + r + 8 * half];
    mrun[r] = -1e30f;
    srun[r] = 0.f;
  }

  // pass 1: online max + rescaled exp-sum (per-lane partials within 16-lane half)
  for (int j0 = 0; j0 < TT; j0 += 16) {
    v8f sc = score_tile(qh + (size_t)(j0 + lid) * DD + koff, afrag);
    const int j = j0 + lid;
#pragma unroll
    for (int r = 0; r < 8; ++r) {
      const int i = i0 + r + 8 * half;
      const float s = bias_elem(sc[r], i, j, inv_s2, gl, gg, er[r], mean, invstd);
      float tmax = s;
#pragma unroll
      for (int mk = 1; mk < 16; mk <<= 1) tmax = fmaxf(tmax, __shfl_xor(tmax, mk, 32));
      const float mnew = fmaxf(mrun[r], tmax);
      srun[r] = srun[r] * __expf(mrun[r] - mnew) + __expf(s - mnew);
      mrun[r] = mnew;
    }
  }
  float rinv[8];
#pragma unroll
  for (int r = 0; r < 8; ++r) {
    float t = srun[r];
#pragma unroll
    for (int mk = 1; mk < 16; mk <<= 1) t += __shfl_xor(t, mk, 32);
    rinv[r] = 1.f / t;
  }

  // pass 2: recompute scores, emit attn, accumulate attn @ v with WMMA.
  __shared__ float tile[16][17];  // C-layout -> A-layout bounce (pad vs bank conflicts)
  v8f accv[4] = {v8f{0.f}, v8f{0.f}, v8f{0.f}, v8f{0.f}};
  float* attn_h = attn + (size_t)hh * TT * TT;

  for (int j0 = 0; j0 < TT; j0 += 16) {
    v8f sc = score_tile(qh + (size_t)(j0 + lid) * DD + koff, afrag);

    // issue attn@v B loads early: independent of scores, latency hidden by
    // the transcendental-heavy softmax epilogue below.
    v2f bv[4][4];
#pragma unroll
    for (int ng = 0; ng < 4; ++ng)
#pragma unroll
      for (int kk = 0; kk < 4; ++kk)
        bv[ng][kk] =
            *(const v2f*)(vh + (size_t)(ng * 16 + lid) * TT + j0 + kk * 4 + koff);

    const int j = j0 + lid;
#pragma unroll
    for (int r = 0; r < 8; ++r) {
      const int i = i0 + r + 8 * half;
      const float s = bias_elem(sc[r], i, j, inv_s2, gl, gg, er[r], mean, invstd);
      const float p = __expf(s - mrun[r]) * rinv[r];
      attn_h[(size_t)i * TT + j] = p;
      tile[r + 8 * half][lid] = p;
    }
    __syncthreads();  // single-wave WG: compiles to wait/NOP, orders LDS
    v2f pa[4];
#pragma unroll
    for (int kk = 0; kk < 4; ++kk) {
      pa[kk].x = tile[lid][kk * 4 + koff];
      pa[kk].y = tile[lid][kk * 4 + koff + 1];
    }
#pragma unroll
    for (int kk = 0; kk < 4; ++kk)
#pragma unroll
      for (int ng = 0; ng < 4; ++ng) accv[ng] = wmma_f32(pa[kk], bv[ng][kk], accv[ng]);
    __syncthreads();
  }
#pragma unroll
  for (int ng = 0; ng < 4; ++ng)
#pragma unroll
    for (int r = 0; r < 8; ++r)
      merged[(size_t)(i0 + r + 8 * half) * EE + hh * DD + ng * 16 + lid] = accv[ng][r];
}

// ---------------------------------------------------------------- launch
extern "C" void kernel_launch(void* const* d_in, const int* in_sizes, int n_in,
                              void* d_out, int out_size, void* d_ws, size_t ws_size,
                              hipStream_t stream) {
  (void)in_sizes; (void)n_in; (void)out_size; (void)ws_size;
  const float* x      = (const float*)d_in[0];
  const float* emo    = (const float*)d_in[1];
  const float* w_qk   = (const float*)d_in[2];
  const float* w_v    = (const float*)d_in[3];
  const float* w_out  = (const float*)d_in[4];
  const float* rms_w  = (const float*)d_in[5];
  const float* sigma  = (const float*)d_in[6];
  const float* beta_l = (const float*)d_in[7];
  const float* beta_g = (const float*)d_in[8];

  float* out  = (float*)d_out;                 // [T,E]
  float* attn = out + (size_t)TT * EE;         // [H,T,T]

  float* ws     = (float*)d_ws;
  float* hbuf   = ws;                          // T*E
  float* shared = hbuf + (size_t)TT * EE;      // T*E
  float* values = shared + (size_t)TT * EE;    // T*E
  float* qn     = values + (size_t)TT * EE;    // H*T*D = T*E
  float* vT     = qn + (size_t)TT * EE;        // H*D*T = T*E
  float* merged = vT + (size_t)TT * EE;        // T*E
  float* partS  = merged + (size_t)TT * EE;    // T
  float* partQ  = partS + TT;                  // T
  float* stats  = partQ + TT;                  // 2

  k_rmsnorm<<<TT, 256, 0, stream>>>(x, rms_w, hbuf);
  k_gemm_nt<<<dim3(EE / 64, TT / 16), 32, 0, stream>>>(hbuf, w_qk, shared, TT, EE, EE);
  k_gemm_nt<<<dim3(EE / 64, TT / 16), 32, 0, stream>>>(hbuf, w_v, values, TT, EE, EE);
  k_headprep<<<dim3(TT, HH), 32, 0, stream>>>(shared, values, qn, vT);
  k_emo_partial<<<TT / 256, 256, 0, stream>>>(emo, partS, partQ);
  k_emo_reduce<<<1, 256, 0, stream>>>(partS, partQ, stats);
  k_attn<<<dim3(TT / 16, HH), 32, 0, stream>>>(qn, vT, emo, stats, sigma, beta_l,
                                               beta_g, attn, merged);
  k_gemm_nt<<<dim3(EE / 64, TT / 16), 32, 0, stream>>>(merged, w_out, out, TT, EE, EE);
}